// SGEncoder_44985487458739
// MI455X (gfx1250) — compile-verified
//
#include <hip/hip_runtime.h>

#define D   40   // feature dim, fixed by the reference
#define DP  48   // D padded to 3 full 16-wide column tiles

typedef __attribute__((ext_vector_type(2))) float v2f;
typedef __attribute__((ext_vector_type(8))) float v8f;

// ---------------------------------------------------------------------------
// deg[i] = 1 (self-loop); then atomic count of dst occurrences; then rsqrt.
// ---------------------------------------------------------------------------
__global__ void init_deg_kernel(float* __restrict__ deg, int n) {
    int i = blockIdx.x * blockDim.x + threadIdx.x;
    if (i < n) deg[i] = 1.0f;  // self-loop contributes 1 to in-degree
}

__global__ void count_deg_kernel(const long long* __restrict__ dst,
                                 float* __restrict__ deg, int E) {
    int e = blockIdx.x * blockDim.x + threadIdx.x;
    if (e < E) atomicAdd(&deg[(int)dst[e]], 1.0f);
}

__global__ void rsqrt_kernel(float* __restrict__ deg, int n) {
    int i = blockIdx.x * blockDim.x + threadIdx.x;
    if (i < n) {
        float d = deg[i];
        deg[i] = (d > 0.0f) ? rsqrtf(d) : 0.0f;  // deg >= 1 here, keep ref semantics
    }
}

// ---------------------------------------------------------------------------
// agg[i,:] = dinv[i]^2 * xin[i,:]   (self-loop term folded into init; also
// wipes the 0xAA-poisoned workspace every call -> deterministic)
// ---------------------------------------------------------------------------
__global__ void init_agg_kernel(const float* __restrict__ xin,
                                const float* __restrict__ dinv,
                                float* __restrict__ agg, long total) {
    long i = (long)blockIdx.x * blockDim.x + threadIdx.x;
    if (i < total) {
        int node = (int)(i / D);
        float di = dinv[node];
        agg[i] = di * di * xin[i];
    }
}

// ---------------------------------------------------------------------------
// One wave per edge, one lane per feature (lanes 0..31 cover f=0..31,
// lanes 0..7 also cover f=32..39). Gather is coalesced; scatter is 40
// contiguous global_atomic_add_f32 that resolve in L2 (16 MB agg << 192 MB).
// ---------------------------------------------------------------------------
__global__ void scatter_kernel(const float* __restrict__ xin,
                               const long long* __restrict__ src,
                               const long long* __restrict__ dst,
                               const float* __restrict__ dinv,
                               float* __restrict__ agg, int E) {
    int gid  = blockIdx.x * blockDim.x + threadIdx.x;
    int e    = gid >> 5;               // wave-uniform edge index (wave32)
    int lane = threadIdx.x & 31;
    if (e >= E) return;                // wave-uniform exit
    int s = (int)src[e];
    int d = (int)dst[e];
    float norm = dinv[s] * dinv[d];
    const float* xs = xin + (size_t)s * D;
    float*       ad = agg + (size_t)d * D;
    atomicAdd(&ad[lane], norm * xs[lane]);
    if (lane < D - 32)
        atomicAdd(&ad[32 + lane], norm * xs[32 + lane]);
}

// ---------------------------------------------------------------------------
// out[n,j] = sum_k A[n,k] * W[j,k] + bias[j]  (optionally ReLU), via
// V_WMMA_F32_16X16X4_F32.
//
//  * W (40x40) + bias are staged zero-padded to 48 output columns in LDS once
//    per block -> B fragments are unconditional ds_load_b64, zero branches.
//  * One wave owns one 16-row tile; A row index is clamped (masked at store),
//    so the 10 A fragments are unconditional global_load_b64, preloaded once
//    and reused across all 3 column tiles (30 WMMAs per wave).
//  * Epilogue: wave-uniform full-tile fast path -> a single per-lane
//    (bcol < D) exec region covering all 8 stores, base pointer + immediate
//    offsets. Ragged row tail (never taken for N=100000) keeps full guards.
//
// ISA 7.12.2 f32 layouts:
//   A 16x4 : lanes 0-15 -> M=lane, VGPR{0,1}=K{0,1}; lanes 16-31 -> K{2,3}
//   B 4x16 : lanes index N, VGPRs index K (mirror of A)
//   C/D    : VGPR v, lanes 0-15 -> (M=v, N=lane); lanes 16-31 -> (M=v+8, N)
// ---------------------------------------------------------------------------
template <bool RELU>
__global__ void gemm_wmma_kernel(const float* __restrict__ A,
                                 const float* __restrict__ W,
                                 const float* __restrict__ bias,
                                 float* __restrict__ outp,
                                 int n) {
    __shared__ float sW[DP * D];   // [j_padded][k], rows 40..47 zeroed
    __shared__ float sB[DP];

    int tid = threadIdx.x;         // 0..127 (4 waves)
    for (int i = tid; i < DP * D; i += 128)
        sW[i] = (i < D * D) ? W[i] : 0.0f;   // rows 0..39 alias W row-major
    if (tid < DP)
        sB[tid] = (tid < D) ? bias[tid] : 0.0f;
    __syncthreads();

    int wave     = (blockIdx.x * blockDim.x + tid) >> 5;
    int rowTiles = (n + 15) >> 4;
    if (wave >= rowTiles) return;  // wave-uniform exit (after barrier)

    int lane = tid & 31;
    int half = lane >> 4;          // 0: K{0,1} pair / 1: K{2,3} pair
    int lm   = lane & 15;
    int row0 = wave << 4;
    int arow = min(row0 + lm, n - 1);   // clamp; out-of-range rows masked at store

    // Preload all 10 A fragments (unconditional, 8B-aligned b64 loads).
    v2f a[10];
    const float* ap = A + (size_t)arow * D + half * 2;
#pragma unroll
    for (int k = 0; k < 10; ++k)
        a[k] = *(const v2f*)(ap + 4 * k);

    bool fullTile = (row0 + 16 <= n);   // wave-uniform

#pragma unroll
    for (int jt = 0; jt < 3; ++jt) {
        int j0   = jt << 4;
        int bcol = j0 + lm;                         // < 48, always valid in LDS
        const float* bp = &sW[bcol * D + half * 2]; // even offset -> 8B aligned
        v8f c = {};
#pragma unroll
        for (int k = 0; k < 10; ++k) {
            v2f b = *(const v2f*)(bp + 4 * k);      // ds_load_b64
            // 8 args: (neg_a, A, neg_b, B, c_mod, C, reuse_a, reuse_b)
            c = __builtin_amdgcn_wmma_f32_16x16x4_f32(
                    false, a[k], false, b, (short)0, c, false, false);
        }
        float  bv = sB[bcol];
        float* op = outp + (size_t)(row0 + half * 8) * D + bcol;

        if (fullTile) {
            if (bcol < D) {                 // one exec region for all 8 stores
#pragma unroll
                for (int v = 0; v < 8; ++v) {
                    float o = c[v] + bv;
                    if (RELU) o = fmaxf(o, 0.0f);
                    op[(size_t)v * D] = o;  // immediate offsets v*160B
                }
            }
        } else {                            // ragged tail (generic path)
#pragma unroll
            for (int v = 0; v < 8; ++v) {
                int row = row0 + v + half * 8;
                if (row < n && bcol < D) {
                    float o = c[v] + bv;
                    if (RELU) o = fmaxf(o, 0.0f);
                    op[(size_t)v * D] = o;
                }
            }
        }
    }
}

// ---------------------------------------------------------------------------
// Orchestration: deg -> dinv -> [init_agg -> scatter -> wmma-gemm] x 2
// ---------------------------------------------------------------------------
extern "C" void kernel_launch(void* const* d_in, const int* in_sizes, int n_in,
                              void* d_out, int out_size, void* d_ws, size_t ws_size,
                              hipStream_t stream) {
    const float*     x   = (const float*)d_in[0];
    const long long* ei  = (const long long*)d_in[1];   // int64 edge_index [2, E]
    const float*     W1  = (const float*)d_in[2];
    const float*     b1  = (const float*)d_in[3];
    const float*     W2  = (const float*)d_in[4];
    const float*     b2  = (const float*)d_in[5];

    const int n = in_sizes[0] / D;          // 100000 nodes
    const int E = in_sizes[1] / 2;          // 1.6M edges
    const long long* src = ei;
    const long long* dst = ei + E;

    // workspace: deg/dinv [n] | agg [n*D] | h [n*D]  (~32.4 MB)
    float* deg = (float*)d_ws;
    float* agg = deg + n;
    float* h   = agg + (size_t)n * D;
    float* out = (float*)d_out;

    const int  T   = 256;
    const long nd  = (long)n * D;
    const long sct = (long)E * 32;          // one wave (32 lanes) per edge

    const int rowTiles   = (n + 15) >> 4;
    const int gemmBlocks = (rowTiles + 3) / 4;  // 4 waves (row tiles) per block

    // shared normalization for both layers
    init_deg_kernel <<<(n + T - 1) / T, T, 0, stream>>>(deg, n);
    count_deg_kernel<<<(E + T - 1) / T, T, 0, stream>>>(dst, deg, E);
    rsqrt_kernel    <<<(n + T - 1) / T, T, 0, stream>>>(deg, n);   // deg -> dinv

    // layer 1: h = relu(prop(x) @ W1^T + b1)
    init_agg_kernel<<<(int)((nd + T - 1) / T), T, 0, stream>>>(x, deg, agg, nd);
    scatter_kernel <<<(int)((sct + T - 1) / T), T, 0, stream>>>(x, src, dst, deg, agg, E);
    gemm_wmma_kernel<true><<<gemmBlocks, 128, 0, stream>>>(agg, W1, b1, h, n);

    // layer 2: out = prop(h) @ W2^T + b2
    init_agg_kernel<<<(int)((nd + T - 1) / T), T, 0, stream>>>(h, deg, agg, nd);
    scatter_kernel <<<(int)((sct + T - 1) / T), T, 0, stream>>>(h, src, dst, deg, agg, E);
    gemm_wmma_kernel<false><<<gemmBlocks, 128, 0, stream>>>(agg, W2, b2, out, n);
}